// Involution_4183298146863
// MI455X (gfx1250) — compile-verified
//
#include <hip/hip_runtime.h>
#include <math.h>

typedef __attribute__((ext_vector_type(2))) float v2f;
typedef __attribute__((ext_vector_type(8))) float v8f;

#define HW 3136   // 56*56
#define IMG 56

// ---------------------------------------------------------------------------
// Fused GEMM:  Out[b][m][n] = epilogue( sum_k  Wt[m][k] * Xin[b][k][n] )
//   epilogue: bn_relu=1 -> ReLU(BN(acc + bias))  (eval-mode batchnorm)
//             bn_relu=0 -> acc + bias
// One wave computes a 16(M) x 64(N) tile with V_WMMA_F32_16X16X4_F32 (exact
// fp32). A-fragment: lane<16 holds row m0+lane, K pair (k0,k0+1); lane>=16
// holds K pair (k0+2,k0+3) -> one contiguous float2 load per lane.
// B/D fragments: N = lane%16, rows split at lane 16 per the ISA layout.
// ---------------------------------------------------------------------------
__global__ __launch_bounds__(256)
void wmma_gemm_f32(const float* __restrict__ Wt,   // [M,K]
                   const float* __restrict__ Xin,  // [B][K,N]
                   float* __restrict__ Out,        // [B][M,N]
                   int M, int N, int K,
                   const float* __restrict__ bias,
                   const float* __restrict__ gamma,
                   const float* __restrict__ beta,
                   const float* __restrict__ mean,
                   const float* __restrict__ var,
                   int bn_relu)
{
    const int lane = threadIdx.x & 31;
    const int wave = threadIdx.x >> 5;
    const int mt   = blockIdx.y * 8 + wave;        // wave-uniform -> EXEC stays full
    if (mt * 16 >= M) return;
    const int m0 = mt * 16;
    const int n0 = blockIdx.x * 64;
    const int lm = lane & 15;
    const int hi = lane >> 4;                      // 0: K pair (0,1), 1: K pair (2,3)

    const float* Xb = Xin + (size_t)blockIdx.z * (size_t)K * (size_t)N;
    float*       Ob = Out + (size_t)blockIdx.z * (size_t)M * (size_t)N;

    const float* wrow = Wt + (size_t)(m0 + lm) * K + 2 * hi;
    const float* xcol = Xb + (size_t)(2 * hi) * N + n0 + lm;

    v8f acc0 = {}, acc1 = {}, acc2 = {}, acc3 = {};
    for (int k0 = 0; k0 < K; k0 += 4) {
        v2f a = *(const v2f*)(wrow + k0);
        const float* xp = xcol + (size_t)k0 * N;
        v2f b0, b1, b2, b3;
        b0[0] = xp[0];       b0[1] = xp[N];
        b1[0] = xp[16];      b1[1] = xp[N + 16];
        b2[0] = xp[32];      b2[1] = xp[N + 32];
        b3[0] = xp[48];      b3[1] = xp[N + 48];
        acc0 = __builtin_amdgcn_wmma_f32_16x16x4_f32(false, a, false, b0, (short)0, acc0, false, false);
        acc1 = __builtin_amdgcn_wmma_f32_16x16x4_f32(false, a, false, b1, (short)0, acc1, false, false);
        acc2 = __builtin_amdgcn_wmma_f32_16x16x4_f32(false, a, false, b2, (short)0, acc2, false, false);
        acc3 = __builtin_amdgcn_wmma_f32_16x16x4_f32(false, a, false, b3, (short)0, acc3, false, false);
    }

    const int col = n0 + lm;
    #pragma unroll
    for (int i = 0; i < 8; ++i) {
        const int ch = m0 + i + hi * 8;            // D row for VGPR i, lane half
        float s, t;
        if (bn_relu) {
            s = gamma[ch] * rsqrtf(var[ch] + 1e-5f);
            t = (bias[ch] - mean[ch]) * s + beta[ch];
        } else {
            s = 1.0f;
            t = bias[ch];
        }
        float o0 = fmaf(acc0[i], s, t);
        float o1 = fmaf(acc1[i], s, t);
        float o2 = fmaf(acc2[i], s, t);
        float o3 = fmaf(acc3[i], s, t);
        if (bn_relu) {
            o0 = fmaxf(o0, 0.f); o1 = fmaxf(o1, 0.f);
            o2 = fmaxf(o2, 0.f); o3 = fmaxf(o3, 0.f);
        }
        float* op = Ob + (size_t)ch * N + col;
        op[0] = o0; op[16] = o1; op[32] = o2; op[48] = o3;
    }
}

// ---------------------------------------------------------------------------
// Involution gather: out[b][g*16+gc][y][x] = 49*eps +
//     sum_{ky,kx} wm[b][g*49 + ky*7+kx][y][x] * xpad[b][g*16+gc][y+ky-3][x+kx-3]
// One block per (b, g, y). Weight slab and 16-ch input patch staged in LDS.
// ---------------------------------------------------------------------------
__global__ __launch_bounds__(256)
void involution_gather(const float* __restrict__ x,    // [4,256,56,56]
                       const float* __restrict__ wm,   // [4,784,56,56]
                       float* __restrict__ out)        // [4,256,56,56]
{
    __shared__ float wsm[49 * IMG];       // 10976 B
    __shared__ float xs[16][7][64];       // 28672 B (cols padded to 64)

    const int y   = blockIdx.x;
    const int g   = blockIdx.y;
    const int b   = blockIdx.z;
    const int tid = threadIdx.x;

    // weights: 49 rows of 56 contiguous floats, strided HW apart
    const float* wm_base = wm + ((size_t)b * 784 + (size_t)g * 49) * HW + y * IMG;
    for (int i = tid; i < 49 * IMG; i += 256) {
        int k = i / IMG, xc = i % IMG;
        wsm[i] = wm_base[(size_t)k * HW + xc];
    }

    // input patch: 16 channels x 7 rows x 62 valid cols (zero-padded borders)
    const float* x_base = x + ((size_t)b * 256 + (size_t)g * 16) * HW;
    for (int i = tid; i < 16 * 7 * 64; i += 256) {
        int ch  = i / (7 * 64);
        int rem = i % (7 * 64);
        int ky  = rem / 64, xc = rem % 64;
        int gy  = y + ky - 3, gx = xc - 3;
        float v = 0.f;
        if (gy >= 0 && gy < IMG && gx >= 0 && gx < IMG)
            v = x_base[(size_t)ch * HW + gy * IMG + gx];
        xs[ch][ky][xc] = v;
    }
    __syncthreads();

    float* out_base = out + ((size_t)b * 256 + (size_t)g * 16) * HW + y * IMG;
    const float EPS49 = 49.0f * 1.1920928955078125e-07f;  // reference adds eps per tap
    for (int i = tid; i < 16 * IMG; i += 256) {
        int gc = i / IMG, xc = i % IMG;
        float acc = EPS49;
        #pragma unroll
        for (int ky = 0; ky < 7; ++ky)
            #pragma unroll
            for (int kx = 0; kx < 7; ++kx)
                acc = fmaf(wsm[(ky * 7 + kx) * IMG + xc], xs[gc][ky][xc + kx], acc);
        out_base[(size_t)gc * HW + xc] = acc;
    }
}

// ---------------------------------------------------------------------------
extern "C" void kernel_launch(void* const* d_in, const int* in_sizes, int n_in,
                              void* d_out, int out_size, void* d_ws, size_t ws_size,
                              hipStream_t stream)
{
    const float* x     = (const float*)d_in[0];   // [4,256,56,56]
    const float* w1    = (const float*)d_in[1];   // [128,256]
    const float* b1    = (const float*)d_in[2];   // [128]
    const float* gamma = (const float*)d_in[3];   // [128]
    const float* beta  = (const float*)d_in[4];   // [128]
    const float* mean  = (const float*)d_in[5];   // [128]
    const float* var   = (const float*)d_in[6];   // [128]
    const float* w2    = (const float*)d_in[7];   // [784,128]
    const float* b2    = (const float*)d_in[8];   // [784]
    float* out = (float*)d_out;

    // workspace: r (4*128*3136 f32 = 6.4MB) then wm (4*784*3136 f32 = 39.3MB)
    float* r  = (float*)d_ws;
    float* wm = r + (size_t)4 * 128 * HW;

    // stage 1: r = ReLU(BN(x @ w1^T + b1));  M=128 -> 8 M-tiles = 1 block.y
    dim3 g1(HW / 64, 1, 4);
    wmma_gemm_f32<<<g1, 256, 0, stream>>>(w1, x, r, 128, HW, 256,
                                          b1, gamma, beta, mean, var, 1);

    // stage 2: wm = r @ w2^T + b2;  M=784 -> 49 M-tiles -> 7 block.y of 8 waves
    dim3 g2(HW / 64, (49 + 7) / 8, 4);
    wmma_gemm_f32<<<g2, 256, 0, stream>>>(w2, r, wm, 784, HW, 128,
                                          b2, nullptr, nullptr, nullptr, nullptr, 0);

    // stage 3: dynamic 7x7 gather
    dim3 g3(IMG, 16, 4);
    involution_gather<<<g3, 256, 0, stream>>>(x, wm, out);
}